// MemoryCrossAttention_20761871909658
// MI455X (gfx1250) — compile-verified
//
#include <hip/hip_runtime.h>
#include <math.h>

typedef float v2f __attribute__((ext_vector_type(2)));
typedef float v8f __attribute__((ext_vector_type(8)));

#define LN_EPS 1e-5f
#define NORM_EPS 1e-12f

// ---------------------------------------------------------------------------
// Optional Tensor Data Mover path (gfx1250 TDM). Descriptor layout per
// CDNA5 ISA sec 8.3-8.6. Guarded so absence of the builtin can't break compile.
// ---------------------------------------------------------------------------
#if defined(__has_builtin)
#if __has_builtin(__builtin_amdgcn_tensor_load_to_lds)
#define HAVE_TDM 1
#endif
#endif

#ifdef HAVE_TDM
typedef unsigned int u32x4 __attribute__((ext_vector_type(4)));
typedef int i32x4 __attribute__((ext_vector_type(4)));
typedef int i32x8 __attribute__((ext_vector_type(8)));

// 2D tile load: rows x cols f32 elements, row stride srcStride elements,
// packed contiguously into LDS at lds_dst.
__device__ inline void tdm_load_2d_f32(const float* gsrc, void* lds_dst,
                                       unsigned rows, unsigned cols,
                                       unsigned srcStride) {
  unsigned long long ga = (unsigned long long)gsrc;
  unsigned lds = (unsigned)(unsigned long long)lds_dst;  // low 32b = LDS byte addr
  u32x4 g0;
  g0[0] = 1u;                                   // count=1, user mode
  g0[1] = lds;                                  // lds_addr [63:32]
  g0[2] = (unsigned)(ga & 0xFFFFFFFFu);         // global_addr[31:0]
  g0[3] = (unsigned)((ga >> 32) & 0x01FFFFFFu)  // global_addr[56:32]
          | (2u << 30);                         // type=2 ("image")
  i32x8 g1;
  unsigned td0 = cols, td1 = rows;
  g1[0] = (int)(2u << 16);                           // wg_mask=0, data_size=4B
  g1[1] = (int)((td0 & 0xFFFFu) << 16);              // tensor_dim0[15:0]
  g1[2] = (int)(((td0 >> 16) & 0xFFFFu) | ((td1 & 0xFFFFu) << 16));
  g1[3] = (int)(((td1 >> 16) & 0xFFFFu) | ((cols & 0xFFFFu) << 16));  // tile_dim0
  g1[4] = (int)(rows & 0xFFFFu);                     // tile_dim1, tile_dim2=0
  g1[5] = (int)srcStride;                            // tensor_dim0_stride[31:0]
  g1[6] = 0;                                         // stride hi, dim1_stride lo
  g1[7] = 0;
  i32x4 g2 = {0, 0, 0, 0};
  i32x4 g3 = {0, 0, 0, 0};
#if __has_include(<hip/amd_detail/amd_gfx1250_TDM.h>)
  i32x8 g4 = {0, 0, 0, 0, 0, 0, 0, 0};
  __builtin_amdgcn_tensor_load_to_lds(g0, g1, g2, g3, g4, 0);
#else
  __builtin_amdgcn_tensor_load_to_lds(g0, g1, g2, g3, 0);
#endif
}
#endif  // HAVE_TDM

// ---------------------------------------------------------------------------
// LayerNorm: one block per row of 1024
// ---------------------------------------------------------------------------
__global__ __launch_bounds__(256) void ln_kernel(const float* __restrict__ x,
                                                 const float* __restrict__ gg,
                                                 const float* __restrict__ bb,
                                                 float* __restrict__ h) {
  __shared__ float red1[8];
  __shared__ float red2[8];
  const int D = 1024;
  size_t row = blockIdx.x;
  const float* xr = x + row * D;
  int d0 = threadIdx.x * 4;
  float4 v = *(const float4*)(xr + d0);
  int lane = threadIdx.x & 31, wave = threadIdx.x >> 5;
  float s = v.x + v.y + v.z + v.w;
  float ss = v.x * v.x + v.y * v.y + v.z * v.z + v.w * v.w;
#pragma unroll
  for (int off = 16; off >= 1; off >>= 1) {
    s += __shfl_xor(s, off, 32);
    ss += __shfl_xor(ss, off, 32);
  }
  if (lane == 0) { red1[wave] = s; red2[wave] = ss; }
  __syncthreads();
  float tot = 0.f, tot2 = 0.f;
#pragma unroll
  for (int i = 0; i < 8; ++i) { tot += red1[i]; tot2 += red2[i]; }
  float mu = tot * (1.f / 1024.f);
  float var = tot2 * (1.f / 1024.f) - mu * mu;
  float rstd = rsqrtf(var + LN_EPS);
  float4 go = *(const float4*)(gg + d0);
  float4 bo = *(const float4*)(bb + d0);
  float4 o;
  o.x = (v.x - mu) * rstd * go.x + bo.x;
  o.y = (v.y - mu) * rstd * go.y + bo.y;
  o.z = (v.z - mu) * rstd * go.z + bo.z;
  o.w = (v.w - mu) * rstd * go.w + bo.w;
  *(float4*)(h + row * D + d0) = o;
}

// ---------------------------------------------------------------------------
// q_summary = mean over T of h_in  : grid (D/256, B)
// ---------------------------------------------------------------------------
__global__ __launch_bounds__(256) void colsum_kernel(const float* __restrict__ h,
                                                     float* __restrict__ qsum,
                                                     int T, int D) {
  int b = blockIdx.y;
  int d = blockIdx.x * 256 + threadIdx.x;
  const float* base = h + (size_t)b * T * D + d;
  float s = 0.f;
  for (int t = 0; t < T; ++t) s += base[(size_t)t * D];
  qsum[b * D + d] = s / (float)T;
}

// ---------------------------------------------------------------------------
// qn = l2norm(q_summary[:, :512]) : one block per batch
// ---------------------------------------------------------------------------
__global__ __launch_bounds__(256) void qn_kernel(const float* __restrict__ qsum,
                                                 float* __restrict__ qn) {
  __shared__ float red[8];
  int b = blockIdx.x;
  float v0 = qsum[b * 1024 + threadIdx.x];
  float v1 = qsum[b * 1024 + 256 + threadIdx.x];
  float s = v0 * v0 + v1 * v1;
  int lane = threadIdx.x & 31, wave = threadIdx.x >> 5;
#pragma unroll
  for (int off = 16; off >= 1; off >>= 1) s += __shfl_xor(s, off, 32);
  if (lane == 0) red[wave] = s;
  __syncthreads();
  float tot = 0.f;
#pragma unroll
  for (int i = 0; i < 8; ++i) tot += red[i];
  float inv = 1.f / fmaxf(sqrtf(tot), NORM_EPS);
  qn[b * 512 + threadIdx.x] = v0 * inv;
  qn[b * 512 + 256 + threadIdx.x] = v1 * inv;
}

// ---------------------------------------------------------------------------
// sim[b][m] = qn[b] . (mem_keys[m] / ||mem_keys[m]||) : one wave per row
// ---------------------------------------------------------------------------
__global__ __launch_bounds__(256) void sim_kernel(const float* __restrict__ mem_keys,
                                                  const float* __restrict__ qn,
                                                  float* __restrict__ sim, int M) {
  int wave = threadIdx.x >> 5, lane = threadIdx.x & 31;
  int m = blockIdx.x * 8 + wave;
  if (m >= M) return;
  const float* row = mem_keys + (size_t)m * 512;
  float ss = 0.f, d0 = 0.f, d1 = 0.f, d2 = 0.f, d3 = 0.f;
#pragma unroll
  for (int i = 0; i < 16; ++i) {
    int d = lane + i * 32;  // coalesced across lanes
    float v = row[d];
    ss += v * v;
    d0 += v * qn[0 * 512 + d];
    d1 += v * qn[1 * 512 + d];
    d2 += v * qn[2 * 512 + d];
    d3 += v * qn[3 * 512 + d];
  }
#pragma unroll
  for (int off = 16; off >= 1; off >>= 1) {
    ss += __shfl_xor(ss, off, 32);
    d0 += __shfl_xor(d0, off, 32);
    d1 += __shfl_xor(d1, off, 32);
    d2 += __shfl_xor(d2, off, 32);
    d3 += __shfl_xor(d3, off, 32);
  }
  if (lane == 0) {
    float inv = 1.f / fmaxf(sqrtf(ss), NORM_EPS);
    sim[0 * (size_t)M + m] = d0 * inv;
    sim[1 * (size_t)M + m] = d1 * inv;
    sim[2 * (size_t)M + m] = d2 * inv;
    sim[3 * (size_t)M + m] = d3 * inv;
  }
}

// ---------------------------------------------------------------------------
// top-16 per batch: 16 argmax passes with masking. One block per batch.
// ---------------------------------------------------------------------------
__global__ __launch_bounds__(256) void topk_kernel(const float* __restrict__ sim,
                                                   int* __restrict__ top_idx, int M) {
  __shared__ float svals[256];
  __shared__ int sidx[256];
  __shared__ int chosen[16];
  int b = blockIdx.x;
  const float* s = sim + (size_t)b * M;
  for (int p = 0; p < 16; ++p) {
    float best = -3.4e38f;
    int bi = 0;
    for (int m = threadIdx.x; m < M; m += 256) {
      bool skip = false;
      for (int c = 0; c < p; ++c)
        if (chosen[c] == m) skip = true;
      float v = skip ? -3.4e38f : s[m];
      if (v > best) { best = v; bi = m; }
    }
    svals[threadIdx.x] = best;
    sidx[threadIdx.x] = bi;
    __syncthreads();
    for (int off = 128; off >= 1; off >>= 1) {
      if (threadIdx.x < off) {
        if (svals[threadIdx.x + off] > svals[threadIdx.x]) {
          svals[threadIdx.x] = svals[threadIdx.x + off];
          sidx[threadIdx.x] = sidx[threadIdx.x + off];
        }
      }
      __syncthreads();
    }
    if (threadIdx.x == 0) {
      chosen[p] = sidx[0];
      top_idx[b * 16 + p] = sidx[0];
    }
    __syncthreads();
  }
}

// ---------------------------------------------------------------------------
// WMMA f32 GEMM: Out[N x J] = A[N x K] @ W[J x K]^T + bias
// One wave per 32x64 output block: 2 A-fragments x 4 B-fragments -> 8
// V_WMMA_F32_16X16X4_F32 accumulators. 6 x 8B loads feed 8 WMMAs per K-step.
//   gidx      : optional row-gather into A
//   epi == 0  : bias only
//   epi == 1  : exact GELU
//   epi == 2  : Out = xres + gate[row] * (acc + bias)   (final residual)
// Fragment maps per ISA 7.12.2:
//   A 16x4 : lanes 0-15 row M=lane hold K={k,k+1}; lanes 16-31 hold K={k+2,k+3}
//   B 4x16 : lane holds column n=lane%16 == contiguous f32 pair of W's row
//   C/D    : vgpr i -> row i + 8*(lane>=16), col = lane%16
// ---------------------------------------------------------------------------
__global__ __launch_bounds__(256) void gemm_wmma(const float* __restrict__ A, int lda,
                                                 const float* __restrict__ W,
                                                 const float* __restrict__ bias,
                                                 float* __restrict__ Out, int ldo,
                                                 int N, int K, int J,
                                                 const int* __restrict__ gidx, int epi,
                                                 const float* __restrict__ xres,
                                                 const float* __restrict__ gate) {
  const int lane = threadIdx.x & 31;
  const int wave = threadIdx.x >> 5;
  const int wid = blockIdx.x * 8 + wave;
  const int strips = J >> 6;               // 64-col strips
  const int trow = (wid / strips) << 5;    // 32 rows per wave
  const int tcol = (wid % strips) << 6;    // 64 cols per wave
  if (trow >= N) return;

  const int l16 = lane & 15;
  const int hi = lane >> 4;  // 0: K pair {0,1}, 1: K pair {2,3}
  int ar0 = trow + l16, ar1 = trow + 16 + l16;
  if (gidx) { ar0 = gidx[ar0]; ar1 = gidx[ar1]; }
  const float* Arow0 = A + (size_t)ar0 * lda + 2 * hi;
  const float* Arow1 = A + (size_t)ar1 * lda + 2 * hi;
  const float* Wr0 = W + (size_t)(tcol + 0 + l16) * K + 2 * hi;
  const float* Wr1 = W + (size_t)(tcol + 16 + l16) * K + 2 * hi;
  const float* Wr2 = W + (size_t)(tcol + 32 + l16) * K + 2 * hi;
  const float* Wr3 = W + (size_t)(tcol + 48 + l16) * K + 2 * hi;

  v8f acc[2][4];
#pragma unroll
  for (int r = 0; r < 2; ++r)
#pragma unroll
    for (int c = 0; c < 4; ++c) acc[r][c] = (v8f){0.f, 0.f, 0.f, 0.f, 0.f, 0.f, 0.f, 0.f};

  for (int k = 0; k < K; k += 64) {
    __builtin_prefetch(Arow0 + k + 64, 0, 3);
    __builtin_prefetch(Arow1 + k + 64, 0, 3);
    __builtin_prefetch(Wr0 + k + 64, 0, 3);
    __builtin_prefetch(Wr2 + k + 64, 0, 3);
#pragma unroll
    for (int kk = 0; kk < 64; kk += 4) {
      v2f a0 = *(const v2f*)(Arow0 + k + kk);
      v2f a1 = *(const v2f*)(Arow1 + k + kk);
      v2f b0 = *(const v2f*)(Wr0 + k + kk);
      v2f b1 = *(const v2f*)(Wr1 + k + kk);
      v2f b2 = *(const v2f*)(Wr2 + k + kk);
      v2f b3 = *(const v2f*)(Wr3 + k + kk);
      acc[0][0] = __builtin_amdgcn_wmma_f32_16x16x4_f32(false, a0, false, b0, (short)0,
                                                        acc[0][0], false, false);
      acc[0][1] = __builtin_amdgcn_wmma_f32_16x16x4_f32(false, a0, false, b1, (short)0,
                                                        acc[0][1], false, false);
      acc[0][2] = __builtin_amdgcn_wmma_f32_16x16x4_f32(false, a0, false, b2, (short)0,
                                                        acc[0][2], false, false);
      acc[0][3] = __builtin_amdgcn_wmma_f32_16x16x4_f32(false, a0, false, b3, (short)0,
                                                        acc[0][3], false, false);
      acc[1][0] = __builtin_amdgcn_wmma_f32_16x16x4_f32(false, a1, false, b0, (short)0,
                                                        acc[1][0], false, false);
      acc[1][1] = __builtin_amdgcn_wmma_f32_16x16x4_f32(false, a1, false, b1, (short)0,
                                                        acc[1][1], false, false);
      acc[1][2] = __builtin_amdgcn_wmma_f32_16x16x4_f32(false, a1, false, b2, (short)0,
                                                        acc[1][2], false, false);
      acc[1][3] = __builtin_amdgcn_wmma_f32_16x16x4_f32(false, a1, false, b3, (short)0,
                                                        acc[1][3], false, false);
    }
  }

#pragma unroll
  for (int c = 0; c < 4; ++c) {
    float bval = bias ? bias[tcol + c * 16 + l16] : 0.f;
#pragma unroll
    for (int r = 0; r < 2; ++r) {
#pragma unroll
      for (int i = 0; i < 8; ++i) {
        int rr = trow + r * 16 + i + hi * 8;
        size_t off = (size_t)rr * ldo + tcol + c * 16 + l16;
        float v = acc[r][c][i] + bval;
        if (epi == 1) {
          v = 0.5f * v * (1.0f + erff(v * 0.70710678118654752f));
        } else if (epi == 2) {
          v = xres[off] + gate[rr] * v;
        }
        Out[off] = v;
      }
    }
  }
}

// ---------------------------------------------------------------------------
// Attention: per (b,h) the 16 K/V rows staged in LDS (TDM when available);
// wave-per-query-row. Writes result in place over Qp (row consumed first).
// ---------------------------------------------------------------------------
__global__ __launch_bounds__(256) void attn_kernel(float* __restrict__ Qp,
                                                   const float* __restrict__ Kp,
                                                   const float* __restrict__ Vp, int T) {
  __shared__ float Kh[16][128];
  __shared__ float Vh[16][128];
  int bh = blockIdx.x;
  int b = bh >> 3, h = bh & 7;
#ifdef HAVE_TDM
  if ((threadIdx.x >> 5) == 0) {  // wave 0 issues both 16x128 f32 tile DMAs
    const float* ksrc = Kp + (size_t)(b * 16) * 1024 + h * 128;
    const float* vsrc = Vp + (size_t)(b * 16) * 1024 + h * 128;
    tdm_load_2d_f32(ksrc, &Kh[0][0], 16, 128, 1024);
    tdm_load_2d_f32(vsrc, &Vh[0][0], 16, 128, 1024);
    __builtin_amdgcn_s_wait_tensorcnt(0);
  }
#else
  for (int i = threadIdx.x; i < 16 * 128; i += 256) {
    int j = i >> 7, d = i & 127;
    size_t src = (size_t)(b * 16 + j) * 1024 + h * 128 + d;
    Kh[j][d] = Kp[src];
    Vh[j][d] = Vp[src];
  }
#endif
  __syncthreads();
  int wave = threadIdx.x >> 5, lane = threadIdx.x & 31;
  int t = blockIdx.y * 8 + wave;
  if (t >= T) return;
  float* q = Qp + (size_t)(b * T + t) * 1024 + h * 128 + lane * 4;
  float q0 = q[0], q1 = q[1], q2 = q[2], q3 = q[3];
  float p[16];
#pragma unroll
  for (int j = 0; j < 16; ++j) {
    const float* kr = &Kh[j][lane * 4];
    p[j] = q0 * kr[0] + q1 * kr[1] + q2 * kr[2] + q3 * kr[3];
  }
#pragma unroll
  for (int off = 16; off >= 1; off >>= 1) {
#pragma unroll
    for (int j = 0; j < 16; ++j) p[j] += __shfl_xor(p[j], off, 32);
  }
  const float scale = 0.08838834764831845f;  // 1/sqrt(128)
  float mx = -3.4e38f;
#pragma unroll
  for (int j = 0; j < 16; ++j) {
    p[j] *= scale;
    mx = fmaxf(mx, p[j]);
  }
  float sum = 0.f;
#pragma unroll
  for (int j = 0; j < 16; ++j) {
    p[j] = expf(p[j] - mx);
    sum += p[j];
  }
  float inv = 1.f / sum;
  float o0 = 0.f, o1 = 0.f, o2 = 0.f, o3 = 0.f;
#pragma unroll
  for (int j = 0; j < 16; ++j) {
    float a = p[j] * inv;
    const float* vr = &Vh[j][lane * 4];
    o0 += a * vr[0];
    o1 += a * vr[1];
    o2 += a * vr[2];
    o3 += a * vr[3];
  }
  q[0] = o0; q[1] = o1; q[2] = o2; q[3] = o3;
}

// ---------------------------------------------------------------------------
// gate[r] = sigmoid(g1out[r] . g2W + g2b) : one wave per row
// ---------------------------------------------------------------------------
__global__ __launch_bounds__(256) void gate_kernel(const float* __restrict__ g1o,
                                                   const float* __restrict__ g2w,
                                                   const float* __restrict__ g2b,
                                                   float* __restrict__ gate, int NR) {
  int wave = threadIdx.x >> 5, lane = threadIdx.x & 31;
  int r = blockIdx.x * 8 + wave;
  if (r >= NR) return;
  const float* row = g1o + (size_t)r * 512;
  float s = 0.f;
#pragma unroll
  for (int i = 0; i < 16; ++i) {
    int d = lane + i * 32;
    s += row[d] * g2w[d];
  }
#pragma unroll
  for (int off = 16; off >= 1; off >>= 1) s += __shfl_xor(s, off, 32);
  if (lane == 0) gate[r] = 1.f / (1.f + expf(-(s + g2b[0])));
}

// ---------------------------------------------------------------------------
extern "C" void kernel_launch(void* const* d_in, const int* in_sizes, int n_in,
                              void* d_out, int out_size, void* d_ws, size_t ws_size,
                              hipStream_t stream) {
  const float* x        = (const float*)d_in[0];
  const float* mem_keys = (const float*)d_in[1];
  const float* mem_vals = (const float*)d_in[2];
  const float* ln_g     = (const float*)d_in[3];
  const float* ln_b     = (const float*)d_in[4];
  const float* qW       = (const float*)d_in[5];
  const float* qb       = (const float*)d_in[6];
  const float* kW       = (const float*)d_in[7];
  const float* kb       = (const float*)d_in[8];
  const float* vW       = (const float*)d_in[9];
  const float* vb       = (const float*)d_in[10];
  const float* oW       = (const float*)d_in[11];
  const float* ob       = (const float*)d_in[12];
  const float* g1W      = (const float*)d_in[13];
  const float* g1b      = (const float*)d_in[14];
  const float* g2W      = (const float*)d_in[15];
  const float* g2b      = (const float*)d_in[16];

  const int B = 4, T = 2048, D = 1024, DM = 512, M = 100000, Hh = 8;
  const int NR = B * T;  // 8192 rows

  float* ws   = (float*)d_ws;
  float* h_in = ws;                         // NR*D
  float* Qp   = h_in + (size_t)NR * D;      // NR*D (reused as attn out)
  float* g1o  = Qp + (size_t)NR * D;        // NR*512
  float* sim  = g1o + (size_t)NR * 512;     // B*M
  float* qsum = sim + (size_t)B * M;        // B*D
  float* qn   = qsum + B * D;               // B*DM
  float* Kp   = qn + B * DM;                // 64*D
  float* Vp   = Kp + 64 * D;                // 64*D
  float* gate = Vp + 64 * D;                // NR
  int* tidx   = (int*)(gate + NR);          // 64 ints

  ln_kernel<<<NR, 256, 0, stream>>>(x, ln_g, ln_b, h_in);
  colsum_kernel<<<dim3(D / 256, B), 256, 0, stream>>>(h_in, qsum, T, D);
  qn_kernel<<<B, 256, 0, stream>>>(qsum, qn);
  sim_kernel<<<(M + 7) / 8, 256, 0, stream>>>(mem_keys, qn, sim, M);
  topk_kernel<<<B, 256, 0, stream>>>(sim, tidx, M);

  // Q = h_in @ qW^T + qb                     (8192x1024x1024)
  gemm_wmma<<<(NR / 32) * (D / 64) / 8, 256, 0, stream>>>(
      h_in, D, qW, qb, Qp, D, NR, D, D, nullptr, 0, nullptr, nullptr);
  // K = mem_keys[top_idx] @ kW^T + kb        (64x512x1024, gathered)
  gemm_wmma<<<(64 / 32) * (D / 64) / 8, 256, 0, stream>>>(
      mem_keys, DM, kW, kb, Kp, D, 64, DM, D, tidx, 0, nullptr, nullptr);
  // V = mem_vals[top_idx] @ vW^T + vb
  gemm_wmma<<<(64 / 32) * (D / 64) / 8, 256, 0, stream>>>(
      mem_vals, DM, vW, vb, Vp, D, 64, DM, D, tidx, 0, nullptr, nullptr);

  attn_kernel<<<dim3(B * Hh, T / 8), 256, 0, stream>>>(Qp, Kp, Vp, T);

  // g1 = gelu(h_in @ g1W^T + g1b)            (8192x1024x512)
  gemm_wmma<<<(NR / 32) * (512 / 64) / 8, 256, 0, stream>>>(
      h_in, D, g1W, g1b, g1o, 512, NR, D, 512, nullptr, 1, nullptr, nullptr);
  gate_kernel<<<NR / 8, 256, 0, stream>>>(g1o, g2W, g2b, gate, NR);

  // out = x + gate * (attn @ oW^T + ob)      (8192x1024x1024) -> d_out
  gemm_wmma<<<(NR / 32) * (D / 64) / 8, 256, 0, stream>>>(
      Qp, D, oW, ob, (float*)d_out, D, NR, D, D, nullptr, 2, x, gate);
}